// MissModel_15564961481514
// MI455X (gfx1250) — compile-verified
//
#include <hip/hip_runtime.h>

// Fused 20-layer Linear(64,64) chain over 524288 tokens, fp32, using
// V_WMMA_F32_16X16X4_F32. Memory-bound workload (~256 MB HBM traffic at
// 23.3 TB/s => ~11us floor); all intermediate activations stay in
// VGPRs/LDS, weights stream from L2.

typedef __attribute__((ext_vector_type(2))) float v2f;
typedef __attribute__((ext_vector_type(4))) float v4f;
typedef __attribute__((ext_vector_type(8))) float v8f;

#define N_TOK 524288
#define DFEAT 64
#define NLAYER 20
#define ROWS_PER_BLOCK 128   // 8 waves x 16 rows
#define THREADS 256

__global__ __launch_bounds__(THREADS)
void mlp_chain_wmma_kernel(const float* __restrict__ x,
                           const float* __restrict__ W,
                           const float* __restrict__ bias,
                           float* __restrict__ out)
{
    // 32 KB activation staging (wave-private 16x64 slabs), 16 KB weights, bias.
    __shared__ __align__(16) float lds_h[ROWS_PER_BLOCK * DFEAT];
    __shared__ __align__(16) float lds_w[DFEAT * DFEAT];
    __shared__ __align__(16) float lds_b[DFEAT];

    const int tid  = threadIdx.x;
    const int lane = tid & 31;
    const int wave = tid >> 5;
    const int lrow = lane & 15;   // M (A/C row) or N (B/C col) within 16-tile
    const int hi   = lane >> 4;   // lane-half selector in fragment layouts

    const long long blockRow = (long long)blockIdx.x * ROWS_PER_BLOCK;

    // ---- stage x tile into LDS, fully coalesced b128 loads ----
    {
        const v4f* gx = (const v4f*)(x + blockRow * DFEAT);
        v4f* sh = (v4f*)lds_h;
#pragma unroll
        for (int i = 0; i < (ROWS_PER_BLOCK * DFEAT / 4) / THREADS; ++i)
            sh[tid + i * THREADS] = gx[tid + i * THREADS];
    }

    // ---- prologue: fetch layer-0 weights into registers ----
    v4f  wreg[4];
    float breg;
    {
        const v4f* gw = (const v4f*)(W);
#pragma unroll
        for (int i = 0; i < 4; ++i) wreg[i] = gw[tid + i * THREADS];
        breg = (tid < DFEAT) ? bias[tid] : 0.0f;
    }

    __syncthreads();   // x tile staged

    // ---- initial A fragments: h[16 x 64] as 16 fragments of 16x4 fp32 ----
    // A layout (32-bit 16x4): lane = m + 16*(k>=2), vgpr = k%2
    // => lane L holds row (L%16), k-pair {2*(L/16), 2*(L/16)+1} per chunk.
    const int waveBase = wave * 16 * DFEAT;
    v2f afrag[16];
#pragma unroll
    for (int kc = 0; kc < 16; ++kc)
        afrag[kc] = *(const v2f*)&lds_h[waveBase + lrow * DFEAT + kc * 4 + 2 * hi];

    // commit layer-0 weights to LDS
    {
        v4f* sw = (v4f*)lds_w;
#pragma unroll
        for (int i = 0; i < 4; ++i) sw[tid + i * THREADS] = wreg[i];
        if (tid < DFEAT) lds_b[tid] = breg;
    }
    __syncthreads();   // weights visible

    for (int l = 0; l < NLAYER; ++l) {
        // prefetch next layer's weights into registers; latency hides under WMMA
        if (l + 1 < NLAYER) {
            const v4f* gw = (const v4f*)(W + (long long)(l + 1) * DFEAT * DFEAT);
#pragma unroll
            for (int i = 0; i < 4; ++i) wreg[i] = gw[tid + i * THREADS];
            if (tid < DFEAT) breg = bias[(l + 1) * DFEAT + tid];
        }

        const bool last = (l == NLAYER - 1);
        float* gout = out + (blockRow + wave * 16) * DFEAT;

        // y[16x64] = h[16x64] @ W^T[64x64] + b : 4 N-tiles x 16 K-steps
#pragma unroll
        for (int nt = 0; nt < 4; ++nt) {
            v8f acc = {};
#pragma unroll
            for (int kc = 0; kc < 16; ++kc) {
                // B fragment (4x16 of W^T): lane = n + 16*(k>=2), vgpr = k%2
                // B[k][n] = W[n_global][k_global] -> contiguous float2 in W row.
                v2f bfrag = *(const v2f*)
                    &lds_w[(nt * 16 + lrow) * DFEAT + kc * 4 + 2 * hi];
                acc = __builtin_amdgcn_wmma_f32_16x16x4_f32(
                        false, afrag[kc], false, bfrag,
                        (short)0, acc, false, false);
            }
            const float bv = lds_b[nt * 16 + lrow];
#pragma unroll
            for (int r = 0; r < 8; ++r) acc[r] += bv;

            // C layout: vgpr r, lane L -> (M = r + 8*(L/16), N = L%16)
            if (last) {
#pragma unroll
                for (int r = 0; r < 8; ++r)
                    gout[(r + 8 * hi) * DFEAT + nt * 16 + lrow] = acc[r];
            } else {
#pragma unroll
                for (int r = 0; r < 8; ++r)
                    lds_h[waveBase + (r + 8 * hi) * DFEAT + nt * 16 + lrow] = acc[r];
            }
        }

        if (!last) {
            // re-form A fragments from wave-private slab (no barrier needed:
            // each wave reads only what it wrote; LDS ops are in-order per wave)
#pragma unroll
            for (int kc = 0; kc < 16; ++kc)
                afrag[kc] = *(const v2f*)
                    &lds_h[waveBase + lrow * DFEAT + kc * 4 + 2 * hi];

            __syncthreads();  // all waves done reading current lds_w
            v4f* sw = (v4f*)lds_w;
#pragma unroll
            for (int i = 0; i < 4; ++i) sw[tid + i * THREADS] = wreg[i];
            if (tid < DFEAT) lds_b[tid] = breg;
            __syncthreads();  // next-layer weights visible
        }
    }
}

extern "C" void kernel_launch(void* const* d_in, const int* in_sizes, int n_in,
                              void* d_out, int out_size, void* d_ws, size_t ws_size,
                              hipStream_t stream) {
    (void)in_sizes; (void)n_in; (void)out_size; (void)d_ws; (void)ws_size;
    const float* x = (const float*)d_in[0];   // [N, 64]
    const float* W = (const float*)d_in[1];   // [20, 64, 64]  (out, in)
    const float* b = (const float*)d_in[2];   // [20, 64]
    float* out = (float*)d_out;               // [N, 64]

    dim3 grid(N_TOK / ROWS_PER_BLOCK);
    mlp_chain_wmma_kernel<<<grid, THREADS, 0, stream>>>(x, W, b, out);
}